// GraphAttentionLayer_59562606460966
// MI455X (gfx1250) — compile-verified
//
#include <hip/hip_runtime.h>
#include <hip/hip_bf16.h>

// GAT layer: B=4, N=4096, F_in=256, F_out=128 (derived from in_sizes at launch).
// adj (268MB int32) is the only large HBM stream; read twice (stats + fused PV).
// H and W are kept packed in WMMA B-fragment layout so hot loops use b128 loads.
// Block-per-tile with 8-way j/k split + LDS tree reduction for occupancy
// (8192 waves instead of 1024 for the WMMA kernels).

#define ALPHA   0.2f
#define NEG_BIG (-9.0e15f)
#define FIN     256
#define FOUT    128

typedef __attribute__((ext_vector_type(2))) float v2f;
typedef __attribute__((ext_vector_type(8))) float v8f;

// Packed B-fragment layout (Hp and Wp): for row-slab s (4 consecutive rows),
//   P[s*512 + lane*16 + c*2 + t1]  ==  M[4*s + (lane>>4)*2 + t1][c*16 + (lane&15)]
// Each lane's 16 floats are its 8 v2f B-fragments for one WMMA k-step
// => one k-step's B operands = 4 x global_load_b128 per lane.

// ---- partial-accumulator dump/add helpers (LDS tree reduction) -------------
__device__ __forceinline__ void dump_acc(float* buf, const v8f* acc, int lane) {
#pragma unroll
  for (int c = 0; c < 8; ++c) {
    float4 q0 = {acc[c][0], acc[c][1], acc[c][2], acc[c][3]};
    float4 q1 = {acc[c][4], acc[c][5], acc[c][6], acc[c][7]};
    *(float4*)(buf + c * 256 + lane * 8)     = q0;
    *(float4*)(buf + c * 256 + lane * 8 + 4) = q1;
  }
}
__device__ __forceinline__ void add_acc(const float* buf, v8f* acc, int lane) {
#pragma unroll
  for (int c = 0; c < 8; ++c) {
    float4 q0 = *(const float4*)(buf + c * 256 + lane * 8);
    float4 q1 = *(const float4*)(buf + c * 256 + lane * 8 + 4);
    acc[c][0] += q0.x; acc[c][1] += q0.y; acc[c][2] += q0.z; acc[c][3] += q0.w;
    acc[c][4] += q1.x; acc[c][5] += q1.y; acc[c][6] += q1.z; acc[c][7] += q1.w;
  }
}
// 8 partials -> wave 0's registers; 3 rounds, 32KB LDS
#define TREE_REDUCE(part, acc, wave, lane)                                   \
  do {                                                                       \
    if ((wave) >= 4) dump_acc((part) + ((wave) - 4) * 2048, (acc), (lane));  \
    __syncthreads();                                                         \
    if ((wave) < 4) add_acc((part) + (wave) * 2048, (acc), (lane));          \
    __syncthreads();                                                         \
    if ((wave) >= 2 && (wave) < 4)                                           \
      dump_acc((part) + ((wave) - 2) * 2048, (acc), (lane));                 \
    __syncthreads();                                                         \
    if ((wave) < 2) add_acc((part) + (wave) * 2048, (acc), (lane));          \
    __syncthreads();                                                         \
    if ((wave) == 1) dump_acc((part), (acc), (lane));                        \
    __syncthreads();                                                         \
    if ((wave) == 0) add_acc((part), (acc), (lane));                         \
  } while (0)

// ---------------------------------------------------------------------------
// Kernel 0: repack W [256x128] into B-fragment layout Wp (one-time, 128KB).
// ---------------------------------------------------------------------------
__global__ __launch_bounds__(256) void gat_wpack_kernel(
    const float* __restrict__ W, float* __restrict__ Wp) {
  const int g = blockIdx.x * 256 + threadIdx.x;   // 16384 threads, 2 floats each
  const int s    = g >> 8;                        // k-slab (0..63)
  const int rem  = (g & 255) * 2;                 // offset within 512-float slab
  const int ln   = rem >> 4;                      // packed lane 0..31
  const int c    = (rem & 15) >> 1;               // col block 0..7
  const int kb   = 4 * s + ((ln >> 4) << 1);      // source row (k)
  const int col  = c * 16 + (ln & 15);
  float2 q;
  q.x = W[(kb + 0) * FOUT + col];
  q.y = W[(kb + 1) * FOUT + col];
  *(float2*)(Wp + 2 * g) = q;
}

// ---------------------------------------------------------------------------
// Kernel 1: H = X @ W (f32 WMMA). Block per 16-row tile; wave w takes
// k in [32w, 32w+32); LDS tree reduce; wave 0 stores packed Hp and computes
// l1 = H.a1, l2 = H.a2 from the reduced accumulators.
// ---------------------------------------------------------------------------
__global__ __launch_bounds__(256) void gat_h_kernel(
    const float* __restrict__ X, const float* __restrict__ Wp,
    const float* __restrict__ a1, const float* __restrict__ a2,
    float* __restrict__ Hp, float* __restrict__ l1, float* __restrict__ l2) {
  __shared__ float part[4 * 2048];                // 32KB
  const int lane = threadIdx.x & 31;
  const int wave = threadIdx.x >> 5;
  const int tile = blockIdx.x;

  const long long i0 = (long long)tile * 16;
  const int n16   = lane & 15;
  const int khalf = (lane >> 4) << 1;

  const float* Xrow = X + (i0 + n16) * FIN;

  v8f acc[8] = {};
  const int kbeg = wave * (FIN / 8);
  for (int k0 = kbeg; k0 < kbeg + FIN / 8; k0 += 4) {
    v2f a;
    a.x = Xrow[k0 + khalf];
    a.y = Xrow[k0 + khalf + 1];
    const float4* __restrict__ wp =
        (const float4*)(Wp + (long long)(k0 >> 2) * 512 + lane * 16);
    const float4 q[4] = {wp[0], wp[1], wp[2], wp[3]};
#pragma unroll
    for (int c = 0; c < 8; ++c) {
      v2f bf;
      bf.x = (c & 1) ? q[c >> 1].z : q[c >> 1].x;
      bf.y = (c & 1) ? q[c >> 1].w : q[c >> 1].y;
      acc[c] = __builtin_amdgcn_wmma_f32_16x16x4_f32(
          false, a, false, bf, (short)0, acc[c], false, false);
    }
  }

  TREE_REDUCE(part, acc, wave, lane);
  if (wave != 0) return;

  const int rr = (lane >> 4) * 8;   // C rows: lanes 16-31 hold M = v+8
  // packed Hp stores (b128)
#pragma unroll
  for (int v2i = 0; v2i < 4; ++v2i) {
    const int v = 2 * v2i;
    const long long row = i0 + rr + v;
    const long long s   = row >> 2;
    const int t         = (int)(row & 3);
    float* dst = Hp + s * 512 + (long long)(n16 + 16 * (t >> 1)) * 16;
#pragma unroll
    for (int c2 = 0; c2 < 4; ++c2) {
      const int c = 2 * c2;
      float4 q;
      q.x = acc[c][v];     q.y = acc[c][v + 1];
      q.z = acc[c + 1][v]; q.w = acc[c + 1][v + 1];
      *(float4*)(dst + c * 2) = q;
    }
  }
  // l1/l2 from reduced accumulators
  float a1v[8], a2v[8];
#pragma unroll
  for (int c = 0; c < 8; ++c) { a1v[c] = a1[c * 16 + n16]; a2v[c] = a2[c * 16 + n16]; }
#pragma unroll
  for (int v = 0; v < 8; ++v) {
    float s1 = 0.f, s2 = 0.f;
#pragma unroll
    for (int c = 0; c < 8; ++c) { s1 += acc[c][v] * a1v[c]; s2 += acc[c][v] * a2v[c]; }
#pragma unroll
    for (int off = 1; off < 16; off <<= 1) {
      s1 += __shfl_xor(s1, off, 32);
      s2 += __shfl_xor(s2, off, 32);
    }
    if (n16 == 0) { l1[i0 + rr + v] = s1; l2[i0 + rr + v] = s2; }
  }
}

// ---------------------------------------------------------------------------
// Kernel 2: per-row online max & sum-of-exp (HBM floor: 268MB ~= 11.5us).
// One wave per row, int4/float4 vectorized adj stream.
// ---------------------------------------------------------------------------
__global__ __launch_bounds__(256) void gat_rowstats_kernel(
    const int* __restrict__ adj, const float* __restrict__ l1,
    const float* __restrict__ l2, float* __restrict__ rmax,
    float* __restrict__ rsum, int N, int rows) {
  const int row  = (int)((blockIdx.x * (long long)blockDim.x + threadIdx.x) >> 5);
  const int lane = threadIdx.x & 31;
  if (row >= rows) return;

  const int b = row / N;
  const int* __restrict__ adjrow = adj + (long long)row * N;
  const float l1v = l1[row];
  const float* __restrict__ l2b = l2 + (long long)b * N;

  float m = -3.0e38f, s = 0.0f;
  for (int j = lane * 4; j < N; j += 128) {
    const int4   a4 = *(const int4*)(adjrow + j);
    const float4 l4 = *(const float4*)(l2b + j);
    const int   av[4] = {a4.x, a4.y, a4.z, a4.w};
    const float lv[4] = {l4.x, l4.y, l4.z, l4.w};
#pragma unroll
    for (int t = 0; t < 4; ++t) {
      float logit = l1v + lv[t];
      float e  = logit > 0.f ? logit : ALPHA * logit;
      float me = av[t] > 0 ? e : NEG_BIG;
      if (me > m) { s = s * __expf(m - me) + 1.0f; m = me; }
      else        { s += __expf(me - m); }
    }
  }
#pragma unroll
  for (int off = 16; off > 0; off >>= 1) {
    float om = __shfl_xor(m, off, 32);
    float os = __shfl_xor(s, off, 32);
    float nm = fmaxf(m, om);
    s = s * __expf(m - nm) + os * __expf(om - nm);
    m = nm;
  }
  if (lane == 0) { rmax[row] = m; rsum[row] = s; }
}

// ---------------------------------------------------------------------------
// Kernel 3: out = softmax(mask(lrelu(l1_i + l2_j))) @ H, fused.
// Block per 16-row tile; wave w takes j-tiles w, w+8, ... (adj stream stays
// partitioned). Per j-tile per wave: 4 b64 adj + 4 b64 l2 + 16 b128 Hp + 32 wmma.
// LDS tree reduce; wave 0 writes the 16x128 output tile.
// ---------------------------------------------------------------------------
__global__ __launch_bounds__(256) void gat_attn_kernel(
    const float* __restrict__ Hp, const int* __restrict__ adj,
    const float* __restrict__ l1, const float* __restrict__ l2,
    const float* __restrict__ rmax, const float* __restrict__ rsum,
    float* __restrict__ out, int N) {
  __shared__ float part[4 * 2048];                // 32KB
  const int lane = threadIdx.x & 31;
  const int wave = threadIdx.x >> 5;
  const int tile = blockIdx.x;

  const int tpb = N >> 4;
  const int b   = tile / tpb;
  const int n16   = lane & 15;
  const int khalf = (lane >> 4) << 1;

  const long long row   = (long long)tile * 16 + n16;
  const long long brow0 = (long long)b * N;
  const float l1v   = l1[row];
  const float m     = rmax[row];
  const float inv_s = 1.0f / rsum[row];
  const int*   __restrict__ adjrow = adj + row * (long long)N;
  const float* __restrict__ l2b    = l2 + brow0;

  v8f acc[8] = {};
  for (int j0 = wave * 16; j0 < N; j0 += 128) {
#pragma unroll
    for (int kk = 0; kk < 4; ++kk) {
      const int j = j0 + kk * 4 + khalf;           // even -> 8B aligned
      const int2   av = *(const int2*)(adjrow + j);
      const float2 lv = *(const float2*)(l2b + j);
      v2f a;
      {
        float logit = l1v + lv.x;
        float e  = logit > 0.f ? logit : ALPHA * logit;
        float me = av.x > 0 ? e : NEG_BIG;
        a.x = __expf(me - m) * inv_s;
      }
      {
        float logit = l1v + lv.y;
        float e  = logit > 0.f ? logit : ALPHA * logit;
        float me = av.y > 0 ? e : NEG_BIG;
        a.y = __expf(me - m) * inv_s;
      }
      const long long s = (brow0 + j0 + 4 * kk) >> 2;
      const float4* __restrict__ hp = (const float4*)(Hp + s * 512 + lane * 16);
      const float4 q[4] = {hp[0], hp[1], hp[2], hp[3]};
#pragma unroll
      for (int c = 0; c < 8; ++c) {
        v2f bf;
        bf.x = (c & 1) ? q[c >> 1].z : q[c >> 1].x;
        bf.y = (c & 1) ? q[c >> 1].w : q[c >> 1].y;
        acc[c] = __builtin_amdgcn_wmma_f32_16x16x4_f32(
            false, a, false, bf, (short)0, acc[c], false, false);
      }
    }
  }

  TREE_REDUCE(part, acc, wave, lane);
  if (wave != 0) return;

  const long long i0 = (long long)tile * 16;
  const int rr = (lane >> 4) * 8;
#pragma unroll
  for (int c = 0; c < 8; ++c)
#pragma unroll
    for (int v = 0; v < 8; ++v)
      out[(i0 + rr + v) * FOUT + c * 16 + n16] = acc[c][v];
}

// ---------------------------------------------------------------------------
extern "C" void kernel_launch(void* const* d_in, const int* in_sizes, int n_in,
                              void* d_out, int out_size, void* d_ws, size_t ws_size,
                              hipStream_t stream) {
  const float* X   = (const float*)d_in[0];   // [B,N,256]
  const int*   adj = (const int*)  d_in[1];   // [B,N,N]
  // d_in[2] = nd_flags (unused by reference math)
  const float* W   = (const float*)d_in[3];   // [256,128]
  const float* a1  = (const float*)d_in[4];   // [128]
  const float* a2  = (const float*)d_in[5];   // [128]
  float* outp = (float*)d_out;

  const int Fout = in_sizes[4];               // 128
  const int Fin  = in_sizes[3] / Fout;        // 256
  const int BN   = in_sizes[0] / Fin;         // B*N = 16384
  const int N    = (int)((long long)in_sizes[1] / BN);  // 4096

  // workspace: Hp[BN*Fout] | Wp[Fin*Fout] | l1[BN] | l2[BN] | rmax[BN] | rsum[BN]
  float* Hp   = (float*)d_ws;
  float* Wp   = Hp + (long long)BN * Fout;
  float* l1   = Wp + (long long)Fin * Fout;
  float* l2   = l1 + BN;
  float* rmax = l2 + BN;
  float* rsum = rmax + BN;

  const int ntiles  = BN / 16;                // 1024 blocks, 8 waves each
  const int rblocks = (BN * 32 + 255) / 256;  // one wave per row

  gat_wpack_kernel<<<(Fin * Fout / 2 + 255) / 256, 256, 0, stream>>>(W, Wp);
  gat_h_kernel<<<ntiles, 256, 0, stream>>>(X, Wp, a1, a2, Hp, l1, l2);
  gat_rowstats_kernel<<<rblocks, 256, 0, stream>>>(adj, l1, l2, rmax, rsum, N, BN);
  gat_attn_kernel<<<ntiles, 256, 0, stream>>>(Hp, adj, l1, l2, rmax, rsum, outp, N);
}